// NGMConvLayer_24902220382787
// MI455X (gfx1250) — compile-verified
//
#include <hip/hip_runtime.h>
#include <hip/hip_bf16.h>

// ---------------------------------------------------------------------------
// GCNConv layer for MI455X (gfx1250, wave32).
//   out = x@W_self + b_self + b_conv + segsum_col( dinv[row]*dinv[col] * (x@W_conv)[row] )
// with self loops folded in analytically (dinv[i]^2 term).
//
// GEMM part uses V_WMMA_F32_16X16X4_F32 (fp32-exact, matches fp32 reference).
// Edge part is L2-atomic bound: xw (25.6MB) and out (25.6MB) both live in the
// 192MB L2, so random gathers + f32 atomic adds resolve on-chip.
// ---------------------------------------------------------------------------

typedef __attribute__((ext_vector_type(2))) float v2f;
typedef __attribute__((ext_vector_type(8))) float v8f;

#define D 64  // D_IN == D_OUT == 64

// ---------------- degree / normalization --------------------------------
__global__ void init_deg_kernel(unsigned int* __restrict__ deg, int N) {
    int i = blockIdx.x * blockDim.x + threadIdx.x;
    if (i < N) deg[i] = 1u;  // self loop
}

__global__ void count_deg_kernel(const int* __restrict__ col,
                                 unsigned int* __restrict__ deg, int E) {
    int e = blockIdx.x * blockDim.x + threadIdx.x;
    if (e < E) atomicAdd(&deg[col[e]], 1u);   // global_atomic_add_u32
}

__global__ void deg_to_dinv_kernel(float* __restrict__ buf, int N) {
    int i = blockIdx.x * blockDim.x + threadIdx.x;
    if (i < N) {
        unsigned int d = ((const unsigned int*)buf)[i];
        buf[i] = __frsqrt_rn((float)d);   // deg >= 1 always
    }
}

// ---------------- fused dual GEMM + self-loop + bias ---------------------
// One wave computes a 16x64 output tile for BOTH weight matrices.
// A-frag (16x4 f32, 2 VGPRs): lanes 0-15 -> M=lane, K=k0..k0+1
//                             lanes 16-31 -> M=lane-16, K=k0+2..k0+3
// B-frag (4x16 f32, 2 VGPRs): lanes 0-15 -> N=lane, K=k0..k0+1
//                             lanes 16-31 -> N=lane-16, K=k0+2..k0+3
// C/D (16x16 f32, 8 VGPRs):  vgpr r: lanes 0-15 -> M=r, lanes 16-31 -> M=r+8
__global__ void gcn_gemm_wmma_kernel(const float* __restrict__ x,
                                     const float* __restrict__ Wc,
                                     const float* __restrict__ bc,
                                     const float* __restrict__ Ws,
                                     const float* __restrict__ bs,
                                     const float* __restrict__ dinv,
                                     float* __restrict__ xw,
                                     float* __restrict__ out, int N) {
    const int lane   = threadIdx.x & 31;
    const int wave   = threadIdx.x >> 5;
    const int waveG  = blockIdx.x * (blockDim.x >> 5) + wave;
    const int rowBase = waveG * 16;
    if (rowBase >= N) return;              // wave-uniform: EXEC stays all-1s

    const int laneHi = lane >> 4;          // 0 or 1
    const int laneLo = lane & 15;

    // ---- load A fragments for all K (16 frags of 16x4) ----
    int arow = rowBase + laneLo;
    if (arow >= N) arow = N - 1;           // safe clamp (N%16==0 normally)
    const float* ap = x + (size_t)arow * D + 2 * laneHi;
    v2f a[16];
#pragma unroll
    for (int k = 0; k < 16; ++k) {
        const float2 t = *(const float2*)(ap + 4 * k);   // 8B aligned
        a[k].x = t.x; a[k].y = t.y;
    }

    // ---- 4 column tiles of 16 ----
#pragma unroll
    for (int nt = 0; nt < 4; ++nt) {
        const int n = nt * 16 + laneLo;
        v8f accC = {};   // x @ W_conv
        v8f accS = {};   // x @ W_self
#pragma unroll
        for (int k0 = 0; k0 < D; k0 += 4) {
            const int kk = k0 + 2 * laneHi;
            v2f bC, bS;
            bC.x = Wc[(size_t)kk * D + n];
            bC.y = Wc[(size_t)(kk + 1) * D + n];
            bS.x = Ws[(size_t)kk * D + n];
            bS.y = Ws[(size_t)(kk + 1) * D + n];
            accC = __builtin_amdgcn_wmma_f32_16x16x4_f32(
                false, a[k0 >> 2], false, bC, (short)0, accC, false, false);
            accS = __builtin_amdgcn_wmma_f32_16x16x4_f32(
                false, a[k0 >> 2], false, bS, (short)0, accS, false, false);
        }
        const float bias = bs[nt * 16 + laneLo] + bc[nt * 16 + laneLo];
#pragma unroll
        for (int r = 0; r < 8; ++r) {
            const int row = rowBase + r + 8 * laneHi;
            if (row < N) {
                const size_t idx = (size_t)row * D + nt * 16 + laneLo;
                const float xv = accC[r];
                xw[idx] = xv;
                const float di = dinv[row];
                out[idx] = accS[r] + bias + di * di * xv;  // self-loop message
            }
        }
    }
}

// ---------------- edge scatter (L2-resident atomics) ---------------------
// 16 lanes per edge, each lane: float4 gather from xw + 4x f32 atomic adds.
__global__ void gcn_scatter_kernel(const int* __restrict__ ei,
                                   const float* __restrict__ dinv,
                                   const float* __restrict__ xw,
                                   float* __restrict__ out, int E) {
    const int tid = blockIdx.x * blockDim.x + threadIdx.x;
    const int e = tid >> 4;
    if (e >= E) return;
    const int fg = (tid & 15) << 2;                 // feature group of 4
    const int r = ei[e];
    const int c = ei[E + e];
    const float w = dinv[r] * dinv[c];
    const float4 v = *(const float4*)(xw + (size_t)r * D + fg);
    float* o = out + (size_t)c * D + fg;
#if defined(__HIP_DEVICE_COMPILE__)
    unsafeAtomicAdd(o + 0, w * v.x);   // global_atomic_add_f32 (no CAS loop)
    unsafeAtomicAdd(o + 1, w * v.y);
    unsafeAtomicAdd(o + 2, w * v.z);
    unsafeAtomicAdd(o + 3, w * v.w);
#endif
}

// ---------------------------------------------------------------------------
extern "C" void kernel_launch(void* const* d_in, const int* in_sizes, int n_in,
                              void* d_out, int out_size, void* d_ws, size_t ws_size,
                              hipStream_t stream) {
    const float* x  = (const float*)d_in[0];
    const int*   ei = (const int*)d_in[1];     // [2, E] flat
    const float* Wc = (const float*)d_in[2];
    const float* bc = (const float*)d_in[3];
    const float* Ws = (const float*)d_in[4];
    const float* bs = (const float*)d_in[5];
    float* out = (float*)d_out;

    const int N = in_sizes[0] / D;
    const int E = in_sizes[1] / 2;

    // scratch layout: [dinv: N floats (aliases uint degree)] [xw: N*D floats]
    float* dinv = (float*)d_ws;
    size_t off = ((size_t)N * sizeof(float) + 255) & ~(size_t)255;
    float* xw = (float*)((char*)d_ws + off);

    const int T = 256;
    // 1) degree (self loop = 1) + edge in-degree
    init_deg_kernel<<<(N + T - 1) / T, T, 0, stream>>>((unsigned int*)dinv, N);
    count_deg_kernel<<<(E + T - 1) / T, T, 0, stream>>>(ei + E, (unsigned int*)dinv, E);
    // 2) dinv = rsqrt(deg)
    deg_to_dinv_kernel<<<(N + T - 1) / T, T, 0, stream>>>(dinv, N);
    // 3) fused WMMA GEMMs: xw = x@Wc;  out = x@Ws + bs + bc + dinv^2 * xw
    //    8 waves/block, 16 rows/wave -> 128 rows/block
    gcn_gemm_wmma_kernel<<<(N + 127) / 128, T, 0, stream>>>(
        x, Wc, bc, Ws, bs, dinv, xw, out, N);
    // 4) edge scatter: out[col] += dinv[row]*dinv[col] * xw[row]
    const long long work = (long long)E * 16;
    gcn_scatter_kernel<<<(unsigned)((work + T - 1) / T), T, 0, stream>>>(
        ei, dinv, xw, out, E);
}